// GATConv_19937238188611
// MI455X (gfx1250) — compile-verified
//
#include <hip/hip_runtime.h>
#include <hip/hip_bf16.h>
#include <math.h>

#define N_NODES 100000
#define N_EDGES 1600000
#define H_HEADS 4
#define F_IN    128
#define F_OUT   32
#define HF      (H_HEADS * F_OUT)   // 128
#define SLOPE   0.2f

typedef __attribute__((ext_vector_type(2))) float v2f;
typedef __attribute__((ext_vector_type(8))) float v8f;

__device__ __forceinline__ float lrelu(float v) { return v > 0.f ? v : SLOPE * v; }

// monotone float->uint encoding so float max == uint max
__device__ __forceinline__ unsigned enc_f32(float f) {
    unsigned u = __float_as_uint(f);
    return (u & 0x80000000u) ? ~u : (u | 0x80000000u);
}
__device__ __forceinline__ float dec_f32(unsigned e) {
    return (e & 0x80000000u) ? __uint_as_float(e ^ 0x80000000u) : __uint_as_float(~e);
}

// ---------------------------------------------------------------------------
// 0) init: out = bias (bias folded into accumulator), m_enc = enc(-inf), denom = 0
// ---------------------------------------------------------------------------
__global__ void gat_init(float* __restrict__ out, const float* __restrict__ bias,
                         unsigned* __restrict__ m_enc, float* __restrict__ denom) {
    int idx = blockIdx.x * blockDim.x + threadIdx.x;
    if (idx < N_NODES * HF) out[idx] = bias[idx & (HF - 1)];
    if (idx < N_NODES * H_HEADS) {
        m_enc[idx] = enc_f32(-__builtin_inff());
        denom[idx] = 0.f;
    }
}

// ---------------------------------------------------------------------------
// 0b) repack W [H,128,32] into WMMA B-fragment order:
//     wt[head*2048 + ct*1024 + kq*32 + kh*16 + n] = {W[h][kq*4+kh*2][ct*16+n],
//                                                    W[h][kq*4+kh*2+1][ct*16+n]}
//     so lane (kh*16+n) of a wave does one coalesced b64 load per fragment.
// ---------------------------------------------------------------------------
__global__ void gat_repack_w(const float* __restrict__ w, v2f* __restrict__ wt) {
    int idx = blockIdx.x * blockDim.x + threadIdx.x;      // 4*2*32*2*16 = 8192
    if (idx >= H_HEADS * 2 * 32 * 2 * 16) return;
    int n    = idx & 15;
    int kh   = (idx >> 4) & 1;
    int kq   = (idx >> 5) & 31;
    int ct   = (idx >> 10) & 1;
    int head = idx >> 11;
    int k    = kq * 4 + kh * 2;
    int col  = ct * 16 + n;
    const float* wp = w + (size_t)head * (F_IN * F_OUT) + (size_t)k * F_OUT + col;
    v2f v; v.x = wp[0]; v.y = wp[F_OUT];
    wt[idx] = v;
}

// ---------------------------------------------------------------------------
// 1) h[n,h,o] = sum_k x[n,k] * W[h,k,o]   via V_WMMA_F32_16X16X4_F32
//    block = 256 thr = 8 waves over 32 rows; wave w -> head = w&3, rowhalf = w>>2
//    each wave keeps TWO independent accumulator chains (col tiles 0..15, 16..31)
//    grid = N/32 (100000 % 32 == 0)
// ---------------------------------------------------------------------------
__global__ void __launch_bounds__(256)
gat_gemm_wmma(const float* __restrict__ x, const v2f* __restrict__ wt,
              float* __restrict__ h) {
    const int lane = threadIdx.x & 31;
    const int wave = threadIdx.x >> 5;
    const int head = wave & 3;
    const int rh   = wave >> 2;         // row half of the 32-row block
    const int ln   = lane & 15;         // M for A, N for B/D
    const int kh   = lane >> 4;         // K half-select (0: K=k,k+1  1: K=k+2,k+3)

    const int row0 = blockIdx.x * 32 + rh * 16;
    const float* xr = x + (size_t)(row0 + ln) * F_IN;
    const v2f* bp = wt + head * 2048 + kh * 16 + ln;   // + ct*1024 + kq*32

    v8f c0 = {}, c1 = {};
#pragma unroll
    for (int kq = 0; kq < 32; ++kq) {
        v2f a  = *(const v2f*)(xr + kq * 4 + kh * 2);
        v2f b0 = bp[kq * 32];
        v2f b1 = bp[1024 + kq * 32];
        c0 = __builtin_amdgcn_wmma_f32_16x16x4_f32(false, a, false, b0,
                                                   (short)0, c0, false, false);
        c1 = __builtin_amdgcn_wmma_f32_16x16x4_f32(false, a, false, b1,
                                                   (short)0, c1, false, false);
    }
    // D layout: VGPR r -> (M = r + 8*kh, N = ln)
    float* hp = h + (size_t)(row0 + kh * 8) * HF + head * F_OUT + ln;
#pragma unroll
    for (int r = 0; r < 8; ++r) {
        hp[(size_t)r * HF]      = c0[r];
        hp[(size_t)r * HF + 16] = c1[r];
    }
}

// ---------------------------------------------------------------------------
// 2) alpha_src/dst[n,h] = h[n,h,:] . a[h, 0:32 / 32:64]
// ---------------------------------------------------------------------------
__global__ void gat_alpha(const float* __restrict__ h, const float* __restrict__ a,
                          float* __restrict__ alpha_src, float* __restrict__ alpha_dst) {
    int idx = blockIdx.x * blockDim.x + threadIdx.x;     // n*H + head
    if (idx >= N_NODES * H_HEADS) return;
    int head = idx & (H_HEADS - 1);
    const float* hp = h + (size_t)idx * F_OUT;
    const float* as = a + head * 2 * F_OUT;
    const float* ad = as + F_OUT;
    float s = 0.f, d = 0.f;
#pragma unroll
    for (int o = 0; o < F_OUT; ++o) {
        float v = hp[o];
        s += v * as[o];
        d += v * ad[o];
    }
    alpha_src[idx] = s;
    alpha_dst[idx] = d;
}

// ---------------------------------------------------------------------------
// 3) per-edge segment max of leaky_relu(alpha_src[s]+alpha_dst[d])
// ---------------------------------------------------------------------------
__global__ void gat_edge_max(const float* __restrict__ alpha_src,
                             const float* __restrict__ alpha_dst,
                             const int* __restrict__ esrc, const int* __restrict__ edst,
                             unsigned* __restrict__ m_enc) {
    int e = blockIdx.x * blockDim.x + threadIdx.x;
    if (e >= N_EDGES) return;
    int s = esrc[e], d = edst[e];
    const float4 as = *(const float4*)(alpha_src + (size_t)s * 4);
    const float4 ad = *(const float4*)(alpha_dst + (size_t)d * 4);
    unsigned* mp = m_enc + (size_t)d * 4;
    atomicMax(mp + 0, enc_f32(lrelu(as.x + ad.x)));
    atomicMax(mp + 1, enc_f32(lrelu(as.y + ad.y)));
    atomicMax(mp + 2, enc_f32(lrelu(as.z + ad.z)));
    atomicMax(mp + 3, enc_f32(lrelu(as.w + ad.w)));
}

// ---------------------------------------------------------------------------
// 4) finalize m: decode, non-finite -> 0 (matches jnp.where(isfinite, m, 0))
// ---------------------------------------------------------------------------
__global__ void gat_finalize_max(const unsigned* __restrict__ m_enc,
                                 float* __restrict__ m) {
    int idx = blockIdx.x * blockDim.x + threadIdx.x;
    if (idx >= N_NODES * H_HEADS) return;
    float f = dec_f32(m_enc[idx]);
    if (!isfinite(f)) f = 0.f;
    m[idx] = f;
}

// ---------------------------------------------------------------------------
// 5) denom[d,h] += exp(e - m[d,h])
// ---------------------------------------------------------------------------
__global__ void gat_edge_sum(const float* __restrict__ alpha_src,
                             const float* __restrict__ alpha_dst,
                             const float* __restrict__ m,
                             const int* __restrict__ esrc, const int* __restrict__ edst,
                             float* __restrict__ denom) {
    int e = blockIdx.x * blockDim.x + threadIdx.x;
    if (e >= N_EDGES) return;
    int s = esrc[e], d = edst[e];
    const float4 as = *(const float4*)(alpha_src + (size_t)s * 4);
    const float4 ad = *(const float4*)(alpha_dst + (size_t)d * 4);
    const float4 mv = *(const float4*)(m + (size_t)d * 4);
    float* dp = denom + (size_t)d * 4;
    atomicAdd(dp + 0, expf(lrelu(as.x + ad.x) - mv.x));
    atomicAdd(dp + 1, expf(lrelu(as.y + ad.y) - mv.y));
    atomicAdd(dp + 2, expf(lrelu(as.z + ad.z) - mv.z));
    atomicAdd(dp + 3, expf(lrelu(as.w + ad.w) - mv.w));
}

// ---------------------------------------------------------------------------
// 6) scatter: out[d,h,o] += h[s,h,o] * att.  One wave32 per edge; lane = o.
//    h and out are L2-resident (51 MB each, 192 MB L2).
// ---------------------------------------------------------------------------
__global__ void __launch_bounds__(256)
gat_edge_scatter(const float* __restrict__ h,
                 const float* __restrict__ alpha_src, const float* __restrict__ alpha_dst,
                 const float* __restrict__ m, const float* __restrict__ denom,
                 const int* __restrict__ esrc, const int* __restrict__ edst,
                 float* __restrict__ out) {
    int e = blockIdx.x * (blockDim.x >> 5) + (threadIdx.x >> 5);
    if (e >= N_EDGES) return;
    int lane = threadIdx.x & 31;
    int s = esrc[e], d = edst[e];
    const float* hs = h + (size_t)s * HF;
    float*       op = out + (size_t)d * HF;
#pragma unroll
    for (int i = 0; i < H_HEADS; ++i) {
        float ev  = lrelu(alpha_src[(size_t)s * 4 + i] + alpha_dst[(size_t)d * 4 + i]);
        float att = expf(ev - m[(size_t)d * 4 + i]) / (denom[(size_t)d * 4 + i] + 1e-16f);
        atomicAdd(op + i * F_OUT + lane, hs[i * F_OUT + lane] * att);
    }
}

// ---------------------------------------------------------------------------
extern "C" void kernel_launch(void* const* d_in, const int* in_sizes, int n_in,
                              void* d_out, int out_size, void* d_ws, size_t ws_size,
                              hipStream_t stream) {
    const float* x      = (const float*)d_in[0];   // [N, 128]
    const float* weight = (const float*)d_in[1];   // [H, 128, 32]
    const float* a      = (const float*)d_in[2];   // [H, 64]
    const float* bias   = (const float*)d_in[3];   // [H, 32]
    const int*   esrc   = (const int*)d_in[4];     // [E]
    const int*   edst   = (const int*)d_in[5];     // [E]
    float* out = (float*)d_out;                    // [N, 128]

    // workspace layout (floats), base assumed 256B aligned
    float*    h         = (float*)d_ws;                       // N*128 = 12.8M
    float*    alpha_src = h + (size_t)N_NODES * HF;           // N*4
    float*    alpha_dst = alpha_src + (size_t)N_NODES * H_HEADS;
    unsigned* m_enc     = (unsigned*)(alpha_dst + (size_t)N_NODES * H_HEADS);
    float*    m         = (float*)(m_enc + (size_t)N_NODES * H_HEADS);
    float*    denom     = m + (size_t)N_NODES * H_HEADS;
    v2f*      wt        = (v2f*)(denom + (size_t)N_NODES * H_HEADS);  // 8192 v2f = 64KB

    const int TB = 256;
    // 0) init out=bias, m_enc=-inf, denom=0
    gat_init<<<(N_NODES * HF + TB - 1) / TB, TB, 0, stream>>>(out, bias, m_enc, denom);
    // 0b) repack W into WMMA B-fragment order
    gat_repack_w<<<(H_HEADS * 2 * 32 * 2 * 16 + TB - 1) / TB, TB, 0, stream>>>(weight, wt);
    // 1) WMMA GEMM (dual accumulator chains per wave)
    gat_gemm_wmma<<<N_NODES / 32, TB, 0, stream>>>(x, wt, h);
    // 2) alpha
    gat_alpha<<<(N_NODES * H_HEADS + TB - 1) / TB, TB, 0, stream>>>(h, a, alpha_src, alpha_dst);
    // 3) segment max
    gat_edge_max<<<(N_EDGES + TB - 1) / TB, TB, 0, stream>>>(alpha_src, alpha_dst, esrc, edst, m_enc);
    // 4) finalize max
    gat_finalize_max<<<(N_NODES * H_HEADS + TB - 1) / TB, TB, 0, stream>>>(m_enc, m);
    // 5) denom
    gat_edge_sum<<<(N_EDGES + TB - 1) / TB, TB, 0, stream>>>(alpha_src, alpha_dst, m, esrc, edst, denom);
    // 6) scatter (one wave per edge, 8 waves/block)
    gat_edge_scatter<<<N_EDGES / 8, TB, 0, stream>>>(h, alpha_src, alpha_dst, m, denom, esrc, edst, out);
}